// ReadHead_83159156785505
// MI455X (gfx1250) — compile-verified
//
#include <hip/hip_runtime.h>
#include <hip/hip_bf16.h>
#include <math.h>

// Problem constants (from reference setup_inputs)
#define B_  8
#define N_  2048
#define W_  64
#define R_  4
#define CTRL_ 272          // R*W + R + 3R
#define EPS_ 1e-8f
#define TI 64              // rows of prev_links per block in the streaming kernel

typedef __attribute__((ext_vector_type(2))) float v2f;
typedef __attribute__((ext_vector_type(8))) float v8f;

// ---------------------------------------------------------------------------
// Kernel Z: zero the backward-weights scratch (accumulated via atomics).
// ---------------------------------------------------------------------------
__global__ void zero_kernel(float* __restrict__ p, int n) {
    int i = blockIdx.x * blockDim.x + threadIdx.x;
    if (i < n) p[i] = 0.0f;
}

// ---------------------------------------------------------------------------
// Kernel A: content weights. One block per (b, r).
//   scores = (mem . key) / (|key||mem_row| + eps) * beta ; softmax over N.
// ---------------------------------------------------------------------------
__global__ __launch_bounds__(256)
void content_kernel(const float* __restrict__ memory,
                    const float* __restrict__ controls,
                    float* __restrict__ content) {
    __shared__ float key_lds[W_];
    __shared__ float score_lds[N_];
    __shared__ float red[8];

    const int tid  = threadIdx.x;
    const int lane = tid & 31, wave = tid >> 5;
    const int b = blockIdx.x >> 2;
    const int r = blockIdx.x & 3;
    const float* ctrl = controls + b * CTRL_;

    if (tid < W_) key_lds[tid] = ctrl[r * W_ + tid];
    __syncthreads();

    // key norm (each thread redundantly; 64 MACs, trivial)
    float ksq = 0.0f;
    #pragma unroll 16
    for (int w = 0; w < W_; ++w) ksq += key_lds[w] * key_lds[w];
    const float knorm = sqrtf(ksq);

    // beta = 1 + softplus(c), numerically stable
    const float bc = ctrl[R_ * W_ + r];
    const float beta = 1.0f + (fmaxf(bc, 0.0f) + log1pf(__expf(-fabsf(bc))));

    float lmax = -3.4e38f;
    for (int n = tid; n < N_; n += 256) {
        const float4* mrow = (const float4*)(memory + ((size_t)b * N_ + n) * W_);
        float dot = 0.0f, msq = 0.0f;
        #pragma unroll
        for (int q = 0; q < W_ / 4; ++q) {
            float4 m4 = mrow[q];
            dot += m4.x * key_lds[q * 4 + 0] + m4.y * key_lds[q * 4 + 1]
                 + m4.z * key_lds[q * 4 + 2] + m4.w * key_lds[q * 4 + 3];
            msq += m4.x * m4.x + m4.y * m4.y + m4.z * m4.z + m4.w * m4.w;
        }
        float sc = dot / (knorm * sqrtf(msq) + EPS_) * beta;
        score_lds[n] = sc;
        lmax = fmaxf(lmax, sc);
    }
    // block max
    for (int off = 16; off; off >>= 1) lmax = fmaxf(lmax, __shfl_xor(lmax, off, 32));
    if (lane == 0) red[wave] = lmax;
    __syncthreads();
    float gmax = red[0];
    #pragma unroll
    for (int w = 1; w < 8; ++w) gmax = fmaxf(gmax, red[w]);
    __syncthreads();

    float lsum = 0.0f;
    for (int n = tid; n < N_; n += 256) {
        float e = __expf(score_lds[n] - gmax);
        score_lds[n] = e;
        lsum += e;
    }
    for (int off = 16; off; off >>= 1) lsum += __shfl_xor(lsum, off, 32);
    if (lane == 0) red[wave] = lsum;
    __syncthreads();
    float gsum = 0.0f;
    #pragma unroll
    for (int w = 0; w < 8; ++w) gsum += red[w];
    const float inv = 1.0f / gsum;

    float* crow = content + ((size_t)(b * R_ + r)) * N_;
    for (int n = tid; n < N_; n += 256) crow[n] = score_lds[n] * inv;
}

// ---------------------------------------------------------------------------
// Kernel B: THE bandwidth kernel. Streams prev_links (134 MB) exactly once,
// never materializing `links`. Each block owns TI rows of one batch; all 256
// threads cover the j dimension (8 slots each, stride 256).
//   link_ij = ((1 - w_i - w_j) * L_ij + w_i * p_j) * (i != j)
//   fwd[r,i] = sum_j link*rw[r,j]   (wave shuffles + LDS cross-wave reduce)
//   bwd[r,j] = sum_i link*rw[r,i]   (thread-local over rows, global atomics)
// Diagonal handled by post-loop fixup (keeps inner loop pure FMA).
// ---------------------------------------------------------------------------
__global__ __launch_bounds__(256)
void links_kernel(const float* __restrict__ links,
                  const float* __restrict__ ww,
                  const float* __restrict__ prec,
                  const float* __restrict__ rw,
                  float* __restrict__ fwd_out,
                  float* __restrict__ bwd_out) {
    __shared__ float rwi_lds[R_ * TI];
    __shared__ float wi_lds[TI];
    __shared__ float fwdp[TI * 8 * R_];   // per-row, per-wave partial fwd

    const int tid  = threadIdx.x;
    const int lane = tid & 31, wave = tid >> 5;
    const int blkPerB = N_ / TI;                  // 32
    const int b  = blockIdx.x / blkPerB;
    const int i0 = (blockIdx.x % blkPerB) * TI;

    // cooperative preload: rw over this block's rows, w over this block's rows
    {
        int rr = tid >> 6, row = tid & 63;        // 256 threads == 4*64
        rwi_lds[rr * TI + row] = rw[((size_t)(b * R_ + rr)) * N_ + i0 + row];
    }
    if (tid < TI) wi_lds[tid] = ww[(size_t)b * N_ + i0 + tid];

    // per-thread j-slot constants
    float wj[8], pj[8], rwj[R_][8];
    #pragma unroll
    for (int s = 0; s < 8; ++s) {
        int j = tid + (s << 8);
        wj[s] = ww[(size_t)b * N_ + j];
        pj[s] = prec[(size_t)b * N_ + j];
        #pragma unroll
        for (int r = 0; r < R_; ++r) rwj[r][s] = rw[((size_t)(b * R_ + r)) * N_ + j];
    }
    float bacc[8][R_];
    #pragma unroll
    for (int s = 0; s < 8; ++s)
        #pragma unroll
        for (int r = 0; r < R_; ++r) bacc[s][r] = 0.0f;
    __syncthreads();

    const float* lbase = links + ((size_t)b * N_ + i0) * N_ + tid;
    for (int i = 0; i < TI; ++i) {
        const float* lrow = lbase + (size_t)i * N_;
        if (i + 1 < TI) __builtin_prefetch(lrow + N_, 0, 0);   // global_prefetch_b8

        const float wi = wi_lds[i];
        const float om = 1.0f - wi;
        float rwi[R_];
        #pragma unroll
        for (int r = 0; r < R_; ++r) rwi[r] = rwi_lds[r * TI + i];

        float a_s[8], c_s[8], Ls[8];
        #pragma unroll
        for (int s = 0; s < 8; ++s) {
            a_s[s] = om - wj[s];
            c_s[s] = wi * pj[s];
            Ls[s]  = lrow[s << 8];
        }
        float fw[R_] = {0.0f, 0.0f, 0.0f, 0.0f};
        #pragma unroll
        for (int s = 0; s < 8; ++s) {
            float link = fmaf(a_s[s], Ls[s], c_s[s]);
            #pragma unroll
            for (int r = 0; r < R_; ++r) fw[r] = fmaf(link, rwj[r][s], fw[r]);
            #pragma unroll
            for (int r = 0; r < R_; ++r) bacc[s][r] = fmaf(link, rwi[r], bacc[s][r]);
        }
        // diagonal fixup: j == i0+i lives in slot s = (i0+i-tid)/256 of thread (i0+i)%256
        int d = (i0 + i) - tid;
        if (d >= 0 && (d & 255) == 0) {
            int s = d >> 8;
            float link = fmaf(a_s[s], Ls[s], c_s[s]);
            #pragma unroll
            for (int r = 0; r < R_; ++r) {
                fw[r]      = fmaf(-link, rwj[r][s], fw[r]);
                bacc[s][r] = fmaf(-link, rwi[r], bacc[s][r]);
            }
        }
        // wave-level reduce of fwd partials; cross-wave combined after loop
        #pragma unroll
        for (int r = 0; r < R_; ++r) {
            float v = fw[r];
            v += __shfl_xor(v, 16, 32);
            v += __shfl_xor(v, 8, 32);
            v += __shfl_xor(v, 4, 32);
            v += __shfl_xor(v, 2, 32);
            v += __shfl_xor(v, 1, 32);
            if (lane == 0) fwdp[(i * 8 + wave) * R_ + r] = v;
        }
    }
    __syncthreads();
    {   // 256 outputs == 256 threads: row = tid/4, r = tid%4
        int row = tid >> 2, r = tid & 3;
        float s = 0.0f;
        #pragma unroll
        for (int w = 0; w < 8; ++w) s += fwdp[(row * 8 + w) * R_ + r];
        fwd_out[((size_t)(b * R_ + r)) * N_ + i0 + row] = s;
    }
    #pragma unroll
    for (int s = 0; s < 8; ++s)
        #pragma unroll
        for (int r = 0; r < R_; ++r)
            atomicAdd(&bwd_out[((size_t)(b * R_ + r)) * N_ + tid + (s << 8)], bacc[s][r]);
}

// ---------------------------------------------------------------------------
// Kernel C: gates + read_weights + read_vector via V_WMMA_F32_16X16X4_F32.
// One wave per (batch, 16-wide w-tile). D(16x16) = A(16x4) @ B(4x16) acc'd
// over K = N in steps of 4. A rows 0..3 are the R read heads (4..15 zero).
// A layout (ISA 7.12.2): lane m = lane%16, K = 2*(lane/16)+v. B mirrored.
// D: VGPR r, lanes 0..15 -> row M=r, col N=lane.
// ---------------------------------------------------------------------------
__global__ __launch_bounds__(32)
void readvec_kernel(const float* __restrict__ memory,
                    const float* __restrict__ controls,
                    const float* __restrict__ fwd,
                    const float* __restrict__ bwd,
                    const float* __restrict__ content,
                    float* __restrict__ out) {
    __shared__ float rwc[R_ * N_];        // 32 KB: combined read weights
    const int b    = blockIdx.x;
    const int wt   = blockIdx.y;          // w-tile 0..3
    const int lane = threadIdx.x;

    // gates = softmax(controls[..., RW+R:].reshape(R,3)) — each lane redundantly
    float g0[R_], g1[R_], g2[R_];
    const float* ctrl = controls + b * CTRL_ + (R_ * W_ + R_);
    #pragma unroll
    for (int r = 0; r < R_; ++r) {
        float x0 = ctrl[r * 3 + 0], x1 = ctrl[r * 3 + 1], x2 = ctrl[r * 3 + 2];
        float mx = fmaxf(x0, fmaxf(x1, x2));
        float e0 = __expf(x0 - mx), e1 = __expf(x1 - mx), e2 = __expf(x2 - mx);
        float inv = 1.0f / (e0 + e1 + e2);
        g0[r] = e0 * inv; g1[r] = e1 * inv; g2[r] = e2 * inv;
    }
    // stage combined read weights in LDS
    for (int n = lane; n < N_; n += 32) {
        #pragma unroll
        for (int r = 0; r < R_; ++r) {
            size_t idx = ((size_t)(b * R_ + r)) * N_ + n;
            rwc[r * N_ + n] = g0[r] * fwd[idx] + g1[r] * bwd[idx] + g2[r] * content[idx];
        }
    }
    __syncthreads();

    const int m  = lane & 15;             // A row (read head, 0..3 real)
    const int kh = lane >> 4;             // K half select
    const float* mcol = memory + (size_t)b * N_ * W_ + wt * 16 + m;

    v8f acc = {};
    for (int k0 = 0; k0 < N_; k0 += 4) {
        v2f a, bv;
        #pragma unroll
        for (int v = 0; v < 2; ++v) {
            int k = k0 + kh * 2 + v;
            float av = rwc[(m & 3) * N_ + k];           // clamped index, select below
            a[v]  = (m < R_) ? av : 0.0f;               // pad rows 4..15 with zero
            bv[v] = mcol[(size_t)k * W_];               // memory[b][k][wt*16+m]
        }
        acc = __builtin_amdgcn_wmma_f32_16x16x4_f32(
            /*neg_a=*/false, a, /*neg_b=*/false, bv,
            /*c_mod=*/(short)0, acc, /*reuse_a=*/false, /*reuse_b=*/false);
    }
    if (lane < 16) {
        #pragma unroll
        for (int r = 0; r < R_; ++r)
            out[((size_t)(b * R_ + r)) * W_ + wt * 16 + m] = acc[r];
    }
}

// ---------------------------------------------------------------------------
extern "C" void kernel_launch(void* const* d_in, const int* in_sizes, int n_in,
                              void* d_out, int out_size, void* d_ws, size_t ws_size,
                              hipStream_t stream) {
    const float* memory     = (const float*)d_in[0];   // (B, N, W)
    const float* controls   = (const float*)d_in[1];   // (B, 272)
    const float* write_w    = (const float*)d_in[2];   // (B, N)
    const float* prev_links = (const float*)d_in[3];   // (B, N, N)
    const float* prev_prec  = (const float*)d_in[4];   // (B, N)
    const float* prev_rw    = (const float*)d_in[5];   // (B, R, N)
    float* out = (float*)d_out;                        // (B, R, W)

    const int BRN = B_ * R_ * N_;                      // 65536
    float* fwd     = (float*)d_ws;
    float* bwd     = fwd + BRN;
    float* content = bwd + BRN;

    // 1) zero the atomic-accumulated backward scratch
    zero_kernel<<<(BRN + 255) / 256, 256, 0, stream>>>(bwd, BRN);
    // 2) content-based addressing softmax
    content_kernel<<<B_ * R_, 256, 0, stream>>>(memory, controls, content);
    // 3) fused single-pass stream over prev_links (the 134 MB hot loop)
    links_kernel<<<B_ * (N_ / TI), 256, 0, stream>>>(prev_links, write_w, prev_prec,
                                                     prev_rw, fwd, bwd);
    // 4) gate mix + read_vector GEMM via f32 WMMA
    readvec_kernel<<<dim3(B_, W_ / 16), 32, 0, stream>>>(memory, controls, fwd, bwd,
                                                         content, out);
}